// RawEinsumOnly_62878321214324
// MI455X (gfx1250) — compile-verified
//
#include <hip/hip_runtime.h>
#include <hip/hip_bf16.h>

typedef __attribute__((ext_vector_type(2))) float v2f;
typedef __attribute__((ext_vector_type(8))) float v8f;

// ---------------------------------------------------------------------------
// Kernel A: ibsum[e,k] = sum_n gate_up_proj[0,e,k,n]
// One wave per (e,k) row of 2048 contiguous floats (8 KB). 16384 rows total.
// 2048 blocks x 256 threads (8 waves/block). Streams 134 MB -> dominates time.
// ---------------------------------------------------------------------------
__global__ void moe_rowsum_ib(const float* __restrict__ ib,
                              float* __restrict__ ibsum) {
    const int gwave = (int)((blockIdx.x * blockDim.x + threadIdx.x) >> 5); // 0..16383
    const int lane  = (int)(threadIdx.x & 31);
    const float4* row4 = (const float4*)(ib + (size_t)gwave * 2048);
    float4 acc = make_float4(0.f, 0.f, 0.f, 0.f);
    // 2048 floats = 512 float4; 32 lanes -> 16 float4 per lane, fully coalesced.
#pragma unroll
    for (int i = 0; i < 16; ++i) {
        float4 v = row4[lane + i * 32];
        acc.x += v.x; acc.y += v.y; acc.z += v.z; acc.w += v.w;
    }
    float s = (acc.x + acc.y) + (acc.z + acc.w);
#pragma unroll
    for (int off = 16; off > 0; off >>= 1) s += __shfl_down(s, off, 32);
    if (lane == 0) ibsum[gwave] = s;
}

// ---------------------------------------------------------------------------
// Kernel B: hsum[b,k] = sum_m hidden[0,b,m,k]
// One thread per (b,k); loop over m (stride 2048 floats) stays coalesced
// across adjacent k. 64 blocks x 256 threads = 16384 threads. Streams 16 MB.
// ---------------------------------------------------------------------------
__global__ void moe_colsum_hidden(const float* __restrict__ hidden,
                                  float* __restrict__ hsum) {
    const int idx = (int)(blockIdx.x * blockDim.x + threadIdx.x); // 0..16383
    const int b = idx >> 11;
    const int k = idx & 2047;
    const float* p = hidden + (size_t)b * (256 * 2048) + k;
    float s = 0.f;
#pragma unroll 8
    for (int m = 0; m < 256; ++m) s += p[(size_t)m * 2048];
    hsum[idx] = s;
}

// ---------------------------------------------------------------------------
// Kernel C: out = sum_{b,e} sp[b,e] * (hsum[b,:] . ibsum[e,:])
// Padded 16x16 GEMM, K=2048, via V_WMMA_F32_16X16X4_F32.
//   A[m,k] = (m<8) ? hsum[m,k]  : 0   (mask by multiply; loads use m&7)
//   B[k,n] = (n<8) ? ibsum[n,k] : 0
// 1 block, 4 waves; wave w handles K chunk [w*512, w*512+512) = 128 WMMA steps.
// Fragment layouts per ISA 7.12.2:
//   A 16x4 f32: lane<16 -> M=lane, {V0,V1}={K0,K1}; lane>=16 -> {K2,K3}
//   B 4x16 f32: lane<16 -> N=lane, {V0,V1}={K0,K1}; lane>=16 -> {K2,K3}
//   C 16x16 f32: VGPR r, lane<16 -> (M=r, N=lane); lane>=16 -> (M=r+8, N=lane-16)
// ---------------------------------------------------------------------------
__global__ void moe_combine_wmma(const float* __restrict__ hsum,
                                 const float* __restrict__ ibsum,
                                 const float* __restrict__ sparsity,
                                 float* __restrict__ out) {
    const int wave = (int)(threadIdx.x >> 5);   // 0..3
    const int lane = (int)(threadIdx.x & 31);
    const int half = lane >> 4;                 // K sub-offset 2*half
    const int lid  = lane & 15;                 // M (for A) / N (for B)
    const int rowm = lid & 7;                   // clamped row, always in-bounds
    const float mask = (lid < 8) ? 1.f : 0.f;   // zero-pad rows/cols 8..15

    v8f c = {};
    const int k0 = wave * 512;
    for (int s = 0; s < 128; ++s) {
        const int kb = k0 + s * 4 + 2 * half;
        v2f a, b;
        a[0] = mask * hsum [rowm * 2048 + kb + 0];
        a[1] = mask * hsum [rowm * 2048 + kb + 1];
        b[0] = mask * ibsum[rowm * 2048 + kb + 0];
        b[1] = mask * ibsum[rowm * 2048 + kb + 1];
        // (neg_a, A, neg_b, B, c_mod, C, reuse_a, reuse_b)
        c = __builtin_amdgcn_wmma_f32_16x16x4_f32(
                false, a, false, b, (short)0, c, false, false);
    }

    // Fold against sparsity: sp_mask[b,e] = sparsity[0,b,0,e] -> flat b*16 + e.
    // Valid entries live in lanes 0..7 (N=e), VGPRs 0..7 (M=b).
    float part = 0.f;
    if (half == 0 && lid < 8) {
#pragma unroll
        for (int r = 0; r < 8; ++r) part += c[r] * sparsity[r * 16 + lid];
    }
#pragma unroll
    for (int off = 16; off > 0; off >>= 1) part += __shfl_down(part, off, 32);

    __shared__ float wave_part[4];
    if (lane == 0) wave_part[wave] = part;
    __syncthreads();
    if (threadIdx.x == 0)
        out[0] = (wave_part[0] + wave_part[1]) + (wave_part[2] + wave_part[3]);
}

// ---------------------------------------------------------------------------
// inputs (setup_inputs order): hidden_4d (1,8,256,2048) f32,
//                              sparsity (1,8,2,8) f32,
//                              gate_up_proj (1,8,2048,2048) f32
// output: scalar f32.  d_ws: hsum (64 KB) + ibsum (64 KB).
// ---------------------------------------------------------------------------
extern "C" void kernel_launch(void* const* d_in, const int* in_sizes, int n_in,
                              void* d_out, int out_size, void* d_ws, size_t ws_size,
                              hipStream_t stream) {
    const float* hidden   = (const float*)d_in[0];
    const float* sparsity = (const float*)d_in[1];
    const float* gup      = (const float*)d_in[2];
    float* out = (float*)d_out;

    float* hsum  = (float*)d_ws;          // 8*2048 floats
    float* ibsum = hsum + 8 * 2048;       // 8*2048 floats

    moe_rowsum_ib    <<<2048, 256, 0, stream>>>(gup, ibsum);
    moe_colsum_hidden<<<  64, 256, 0, stream>>>(hidden, hsum);
    moe_combine_wmma <<<   1, 128, 0, stream>>>(hsum, ibsum, sparsity, out);
}